// GCN_87230785781866
// MI455X (gfx1250) — compile-verified
//
#include <hip/hip_runtime.h>
#include <cstdint>

#define N_NODES 118
#define F_IN    256
#define H_DIM   256
#define K2      24
#define N_EDGE  372
#define N_E     490     // edges + self loops
#define LP      120     // LDS row pitch in floats (padded from 118)
#define OC      64      // layer-1 output-channel chunk

#define AS1 __attribute__((address_space(1)))
#define AS3 __attribute__((address_space(3)))

#if defined(__has_builtin)
#if __has_builtin(__builtin_amdgcn_global_load_async_to_lds_b32) && \
    __has_builtin(__builtin_amdgcn_s_wait_asynccnt)
#define USE_ASYNC_EDGE_COPY 1
#endif
#endif

typedef __attribute__((ext_vector_type(2))) float v2f;
typedef __attribute__((ext_vector_type(8))) float v8f;

__device__ __forceinline__ v8f wmma4_f32(v2f a, v2f b, v8f c) {
  // D = A(16x4) * B(4x16) + C, full fp32 (exact vs fp32 reference)
  return __builtin_amdgcn_wmma_f32_16x16x4_f32(false, a, false, b,
                                               (short)0, c, false, false);
}

#ifdef USE_ASYNC_EDGE_COPY
__device__ __forceinline__ void async_copy_b32(const void* g, void* l) {
  __builtin_amdgcn_global_load_async_to_lds_b32(
      (AS1 int*)(uintptr_t)g,
      (AS3 int*)(uint32_t)(uintptr_t)l, 0, 0);
}
#endif

// ---- one-block prep: degrees -> rsqrt -> per-edge norm, plus src/dst lists
__global__ void gcn_prep(const int* __restrict__ edge_index,
                         float* __restrict__ norm,
                         int* __restrict__ src, int* __restrict__ dst) {
  __shared__ float deg[N_NODES];
  __shared__ float dinv[N_NODES];
  int t = threadIdx.x;
  if (t < N_NODES) deg[t] = 0.f;
  __syncthreads();
  int s = 0, d = 0;
  if (t < N_E) {
    s = (t < N_EDGE) ? edge_index[t]          : (t - N_EDGE);
    d = (t < N_EDGE) ? edge_index[N_EDGE + t] : (t - N_EDGE);
    src[t] = s; dst[t] = d;
    atomicAdd(&deg[d], 1.0f);
  }
  __syncthreads();
  if (t < N_NODES) dinv[t] = (deg[t] > 0.f) ? rsqrtf(deg[t]) : 0.f;
  __syncthreads();
  if (t < N_E) norm[t] = dinv[s] * dinv[d];
}

// ---- main: one workgroup per batch element, 8 waves
__launch_bounds__(256)
__global__ void gcn_main(const float* __restrict__ x,
                         const float* __restrict__ W1, const float* __restrict__ b1,
                         const float* __restrict__ W2, const float* __restrict__ b2,
                         const float* __restrict__ Wfc, const float* __restrict__ bfc,
                         const float* __restrict__ norm,
                         const int* __restrict__ src, const int* __restrict__ dst,
                         float* __restrict__ out) {
  extern __shared__ float smem[];
  float* hs    = smem;               // [OC][LP] pre-aggregation h1 chunk
  float* ha    = hs + OC * LP;       // [OC][LP] aggregated (+bias) h1 chunk
  float* h2    = ha + OC * LP;       // [K2][LP] pre-aggregation h2
  float* h2a   = h2 + K2 * LP;       // [K2][LP] aggregated h2
  float* snorm = h2a + K2 * LP;      // [N_E]
  int*   ssrc  = (int*)(snorm + N_E);
  int*   sdst  = ssrc + N_E;

  const int b    = blockIdx.x;
  const int tid  = threadIdx.x;
  const int wave = tid >> 5;
  const int lane = tid & 31;
  const int r16  = lane & 15;        // row/col within 16-wide fragment
  const int hi   = lane >> 4;        // K-half selector per ISA f32 layout

#ifdef USE_ASYNC_EDGE_COPY
  for (int i = tid; i < N_E; i += 256) {
    async_copy_b32(norm + i, snorm + i);
    async_copy_b32(src + i,  ssrc + i);
    async_copy_b32(dst + i,  sdst + i);
  }
  __builtin_amdgcn_s_wait_asynccnt(0);
#else
  for (int i = tid; i < N_E; i += 256) {
    snorm[i] = norm[i]; ssrc[i] = src[i]; sdst[i] = dst[i];
  }
#endif
  for (int i = tid; i < K2 * LP; i += 256) h2a[i] = b2[i / LP];
  __syncthreads();

  const float* xg = x + (size_t)b * (F_IN * N_NODES);

  // per-wave n-tile (constant across the whole kernel)
  const int  nIdx  = wave * 16 + r16;
  const bool nok   = nIdx < N_NODES;
  const int  nB    = nok ? nIdx : 0;
  const float nMask = nok ? 1.f : 0.f;

  v8f c2[2];
  for (int i = 0; i < 2; ++i)
    for (int r = 0; r < 8; ++r) c2[i][r] = 0.f;

  for (int oc = 0; oc < H_DIM / OC; ++oc) {
    // ---- layer-1 GEMM chunk: hs[o][n] = sum_f W1[oc*64+o, f] * x[b, f, n]
    // Each wave owns ONE n-tile and all 4 m-tiles: one strided B (x)
    // fragment feeds 4 WMMAs -> 4x fewer strided vector loads.
    {
      const float* Bcol = xg + (size_t)(2 * hi) * N_NODES + nB;
      const float* Ar[4];
      #pragma unroll
      for (int tm = 0; tm < 4; ++tm)
        Ar[tm] = W1 + (size_t)(oc * OC + tm * 16 + r16) * F_IN + 2 * hi;
      v8f ca[4];
      #pragma unroll
      for (int tm = 0; tm < 4; ++tm)
        for (int r = 0; r < 8; ++r) ca[tm][r] = 0.f;
      #pragma unroll 2
      for (int k = 0; k < F_IN; k += 4) {
        v2f bb; bb.x = Bcol[(size_t)k * N_NODES] * nMask;
                bb.y = Bcol[(size_t)(k + 1) * N_NODES] * nMask;
        #pragma unroll
        for (int tm = 0; tm < 4; ++tm) {
          v2f a; a.x = Ar[tm][k]; a.y = Ar[tm][k + 1];
          ca[tm] = wmma4_f32(a, bb, ca[tm]);
        }
      }
      if (nok) {
        #pragma unroll
        for (int tm = 0; tm < 4; ++tm)
          #pragma unroll
          for (int r = 0; r < 8; ++r)
            hs[(tm * 16 + r + 8 * hi) * LP + nIdx] = ca[tm][r];
      }
    }
    // ---- seed aggregation buffer with bias (pad columns -> 0 so relu keeps 0)
    for (int i = tid; i < OC * LP; i += 256) {
      int n = i % LP;
      ha[i] = (n < N_NODES) ? b1[oc * OC + i / LP] : 0.f;
    }
    __syncthreads();
    // ---- normalized scatter-add over edges (LDS float atomics, row per o)
    for (int i = tid; i < OC * N_E; i += 256) {
      int o = i / N_E, e = i - o * N_E;
      atomicAdd(&ha[o * LP + sdst[e]], snorm[e] * hs[o * LP + ssrc[e]]);
    }
    __syncthreads();
    // ---- layer-2 partial GEMM (relu fused into B load), accumulate in regs.
    // B fragment (LDS) loaded once per k-step, shared by both m-tiles.
    {
      const int  mIdx1  = 16 + r16;
      const float mMask1 = (mIdx1 < K2) ? 1.f : 0.f;
      const float* A0 = W2 + (size_t)r16 * H_DIM + oc * OC + 2 * hi;
      const float* A1 = W2 + (size_t)((mIdx1 < K2) ? mIdx1 : 0) * H_DIM
                           + oc * OC + 2 * hi;
      #pragma unroll 4
      for (int k = 0; k < OC; k += 4) {
        v2f bb; bb.x = fmaxf(ha[(k + 2 * hi) * LP + nB], 0.f);
                bb.y = fmaxf(ha[(k + 2 * hi + 1) * LP + nB], 0.f);
        v2f a0; a0.x = A0[k];          a0.y = A0[k + 1];
        c2[0] = wmma4_f32(a0, bb, c2[0]);
        v2f a1; a1.x = A1[k] * mMask1; a1.y = A1[k + 1] * mMask1;
        c2[1] = wmma4_f32(a1, bb, c2[1]);
      }
    }
    __syncthreads();
  }

  // ---- spill layer-2 pre-aggregation tiles to LDS
  if (nok) {
    #pragma unroll
    for (int i = 0; i < 2; ++i) {
      #pragma unroll
      for (int r = 0; r < 8; ++r) {
        int m = i * 16 + r + 8 * hi;
        if (m < K2) h2[m * LP + nIdx] = c2[i][r];
      }
    }
  }
  __syncthreads();
  // ---- layer-2 graph scatter (+b2 already seeded in h2a)
  for (int i = tid; i < K2 * N_E; i += 256) {
    int k = i / N_E, e = i - k * N_E;
    atomicAdd(&h2a[k * LP + sdst[e]], snorm[e] * h2[k * LP + ssrc[e]]);
  }
  __syncthreads();
  // ---- node-mixing FC + final transpose: out[b][o][k], o<54, k<24
  for (int i = tid; i < 54 * K2; i += 256) {
    int o = i / K2, k = i - o * K2;
    float acc = bfc[o];
    #pragma unroll 2
    for (int n = 0; n < N_NODES; ++n)
      acc += Wfc[o * N_NODES + n] * h2a[k * LP + n];
    out[((size_t)b * 54 + o) * K2 + k] = acc;
  }
}

extern "C" void kernel_launch(void* const* d_in, const int* in_sizes, int n_in,
                              void* d_out, int out_size, void* d_ws, size_t ws_size,
                              hipStream_t stream) {
  const float* x          = (const float*)d_in[0];
  const int*   edge_index = (const int*)  d_in[1];
  const float* W1         = (const float*)d_in[2];
  const float* b1         = (const float*)d_in[3];
  const float* W2         = (const float*)d_in[4];
  const float* b2         = (const float*)d_in[5];
  const float* Wfc        = (const float*)d_in[6];
  const float* bfc        = (const float*)d_in[7];
  float*       out        = (float*)d_out;

  float* norm = (float*)d_ws;                 // 490 floats
  int*   src  = (int*)((float*)d_ws + N_E);   // 490 ints
  int*   dst  = src + N_E;                    // 490 ints

  const int B = in_sizes[0] / (F_IN * N_NODES);

  gcn_prep<<<1, 512, 0, stream>>>(edge_index, norm, src, dst);

  size_t smem_bytes =
      (size_t)(OC * LP * 2 + K2 * LP * 2 + N_E) * sizeof(float) +
      (size_t)(N_E * 2) * sizeof(int);
  gcn_main<<<B, 256, smem_bytes, stream>>>(x, W1, b1, W2, b2, Wfc, bfc,
                                           norm, src, dst, out);
}